// RoIPooling2D_24395414241599
// MI455X (gfx1250) — compile-verified
//
#include <hip/hip_runtime.h>
#include <hip/hip_bf16.h>

// RoIPooling2D via 2D summed-area table (matches reference algorithm).
//   K1: row-wise cumsum of x -> S, 16-wide blocked prefix done as triangular
//       matmul on the CDNA5 matrix unit (v_wmma_f32_16x16x4_f32, 4 chained
//       K-chunks per 16x16 tile), carry propagated per channel via __shfl.
//   K2: column-wise cumsum of S in place (coalesced serial scan).
//   K3: per (roi,bin) 4-corner gather + divide by bin area.
// Everything fp32 to track the reference's fp32 cumsum numerics.

#define FH 512
#define FW 512
#define FC 256
#define OUTH 7
#define OUTW 7

typedef __attribute__((ext_vector_type(2))) float v2f;
typedef __attribute__((ext_vector_type(8))) float v8f;

// ---------------------------------------------------------------------------
// Kernel 1: cumsum along W. One wave = one (row h, 16-channel block) strip.
// Within each 16-w block:  D[M][c] = sum_{w'<=M} X[w'][c]  computed as
// L(16x16, lower-tri ones) * X(16x16) via 4x V_WMMA_F32_16X16X4_F32.
// ---------------------------------------------------------------------------
__global__ void __launch_bounds__(256)
rp_row_scan_wmma(const float* __restrict__ x, float* __restrict__ S) {
    const int lane  = threadIdx.x & 31;
    const int wave  = threadIdx.x >> 5;
    const int strip = blockIdx.x * 8 + wave;        // H * (C/16) = 8192 strips
    const int h     = strip >> 4;                   // / (FC/16)
    const int cb    = (strip & 15) << 4;            // channel block base
    const int half  = lane >> 4;                    // 0: lanes 0-15, 1: 16-31
    const int lrow  = lane & 15;                    // M (A/D) or channel (B/D)

    // Triangular A chunks: A_k[M][kk] = (M >= 4k + kk), laid out per the ISA
    // 16x4 f32 A layout: VGPR e holds K = 2*half + e for row M = lrow.
    v2f a[4];
#pragma unroll
    for (int k = 0; k < 4; ++k) {
        const int k0 = 4 * k + 2 * half;
        a[k].x = (lrow >= k0)     ? 1.0f : 0.0f;
        a[k].y = (lrow >= k0 + 1) ? 1.0f : 0.0f;
    }

    const float* xrow = x + (size_t)h * FW * FC + cb + lrow;
    float*       srow = S + (size_t)h * FW * FC + cb + lrow;

    float carry = 0.0f;                             // running prefix, per channel
    for (int w0 = 0; w0 < FW; w0 += 16) {
        // B operand: B_k[kk][c] = X[w0 + 4k + kk][cb + c], 4x16 row-striped.
        v8f d = {};
#pragma unroll
        for (int k = 0; k < 4; ++k) {
            const float* p = xrow + (size_t)(w0 + 4 * k + 2 * half) * FC;
            v2f b;
            b.x = p[0];
            b.y = p[FC];
            d = __builtin_amdgcn_wmma_f32_16x16x4_f32(
                    false, a[k], false, b, (short)0, d, false, false);
        }

        // Block total per channel = D row 15 = VGPR7 on lanes 16-31.
        const float blocksum = __shfl(d[7], lrow + 16, 32);

        // D layout: VGPR r holds row r (lanes 0-15) / row r+8 (lanes 16-31).
#pragma unroll
        for (int r = 0; r < 8; ++r) {
            const int row = w0 + r + 8 * half;
            srow[(size_t)row * FC] = d[r] + carry;
        }
        carry += blocksum;
    }
}

// ---------------------------------------------------------------------------
// Kernel 2: cumsum along H, in place. Thread t -> (w, c); lanes contiguous in
// c so every access is a full 128B/wave coalesced transaction. Unrolled x4 so
// loads pipeline ahead of the dependent add chain.
// ---------------------------------------------------------------------------
__global__ void __launch_bounds__(256)
rp_col_scan(float* __restrict__ S) {
    const int t = blockIdx.x * 256 + threadIdx.x;   // over W*C
    const int c = t & (FC - 1);
    const int w = t >> 8;
    const size_t stride = (size_t)FW * FC;

    float* p = S + (size_t)w * FC + c;
    float acc = 0.0f;
    for (int h = 0; h < FH; h += 4) {
        const float v0 = p[0];
        const float v1 = p[stride];
        const float v2 = p[2 * stride];
        const float v3 = p[3 * stride];
        acc += v0; p[0]          = acc;
        acc += v1; p[stride]     = acc;
        acc += v2; p[2 * stride] = acc;
        acc += v3; p[3 * stride] = acc;
        p += 4 * stride;
    }
}

// ---------------------------------------------------------------------------
// Kernel 3: one block per (roi, bin); thread = channel. S is the INCLUSIVE
// 2D prefix sum, so corner index -1 contributes 0 (replaces zero padding).
// ---------------------------------------------------------------------------
__global__ void __launch_bounds__(256)
rp_gather(const float* __restrict__ S, const float* __restrict__ rois,
          float* __restrict__ out) {
    const int blk = blockIdx.x;
    const int n   = blk / (OUTH * OUTW);
    const int ij  = blk % (OUTH * OUTW);
    const int i   = ij / OUTW;
    const int j   = ij % OUTW;
    const int c   = threadIdx.x;

    const float y0 = rois[4 * n + 0];
    const float x0 = rois[4 * n + 1];
    const float y1 = rois[4 * n + 2];
    const float x1 = rois[4 * n + 3];

    const int hs = (int)floorf((float)FH * y0);
    const int ws = (int)floorf((float)FW * x0);
    int he = (int)floorf((float)FH * y1);
    int we = (int)floorf((float)FW * x1);
    he = max(he, hs + 1);
    we = max(we, ws + 1);
    const int rh = he - hs;
    const int rw = we - ws;

    // AdaptiveAvgPool2d bin edges (all quantities non-negative).
    const int hb0 = hs + (i * rh) / OUTH;
    const int hb1 = hs + ((i + 1) * rh + OUTH - 1) / OUTH;
    const int wb0 = ws + (j * rw) / OUTW;
    const int wb1 = ws + ((j + 1) * rw + OUTW - 1) / OUTW;

    const int a1 = hb1 - 1, a0 = hb0 - 1;
    const int b1 = wb1 - 1, b0 = wb0 - 1;

    auto g = [&](int a, int b) -> float {
        if (a < 0 || b < 0) return 0.0f;
        return S[((size_t)a * FW + b) * FC + c];
    };

    const float s = g(a1, b1) - g(a0, b1) - g(a1, b0) + g(a0, b0);
    const float area = (float)((hb1 - hb0) * (wb1 - wb0));
    out[(size_t)blk * FC + c] = s / area;
}

// ---------------------------------------------------------------------------
extern "C" void kernel_launch(void* const* d_in, const int* in_sizes, int n_in,
                              void* d_out, int out_size, void* d_ws, size_t ws_size,
                              hipStream_t stream) {
    const float* x    = (const float*)d_in[0];
    const float* rois = (const float*)d_in[1];
    // d_in[2] (spatial_scale) is unused by the reference computation.
    float* out = (float*)d_out;
    float* S   = (float*)d_ws;   // needs FH*FW*FC*4 = 256 MiB

    const int N = in_sizes[1] / 4;

    // K1: H * (C/16) / (waves per block) = 8192 / 8 = 1024 blocks.
    rp_row_scan_wmma<<<1024, 256, 0, stream>>>(x, S);
    // K2: W*C / 256 = 512 blocks.
    rp_col_scan<<<512, 256, 0, stream>>>(S);
    // K3: one block per (roi, bin).
    rp_gather<<<N * OUTH * OUTW, 256, 0, stream>>>(S, rois, out);
}